// ElecFormer_88948772700512
// MI455X (gfx1250) — compile-verified
//
#include <hip/hip_runtime.h>
#include <hip/hip_bf16.h>

#define BGRAPHS 32
#define NNODES  256
#define DNODE   128
#define DPAIR   64
#define EPS_Z   1e-8f
#define EPS_LN  1e-5f

typedef __attribute__((ext_vector_type(2))) float v2f;
typedef __attribute__((ext_vector_type(8))) float v8f;

// ---------------------------------------------------------------------------
// Kernel 1: exact per-graph median of 65536 positive floats via 4-pass
// radix select on IEEE bit patterns (positive floats order like uint32).
// Finds order statistics 32767 and 32768 (0-indexed), averages them.
// One 256-thread block per graph. Bandwidth: 8 passes * 8MB -> negligible.
// ---------------------------------------------------------------------------
__global__ void __launch_bounds__(256)
median_kernel(const float* __restrict__ z, float* __restrict__ med) {
  const int b   = blockIdx.x;
  const int tid = threadIdx.x;
  const unsigned* zb = (const unsigned*)z + (size_t)b * (NNODES * NNODES);

  __shared__ int      hist[256];
  __shared__ unsigned s_prefix;
  __shared__ int      s_k;

  unsigned result[2];
  for (int sel = 0; sel < 2; ++sel) {
    int      k      = (NNODES * NNODES) / 2 - 1 + sel;  // 0-indexed kth smallest
    unsigned prefix = 0u;
    for (int pass = 0; pass < 4; ++pass) {
      const int      shift = 24 - 8 * pass;
      const unsigned mask  = (pass == 0) ? 0u : (0xFFFFFFFFu << (shift + 8));
      hist[tid] = 0;
      __syncthreads();
      for (int i = tid; i < NNODES * NNODES; i += 256) {
        unsigned v = zb[i];
        if ((v & mask) == prefix) atomicAdd(&hist[(v >> shift) & 255], 1);
      }
      __syncthreads();
      if (tid == 0) {
        int acc = 0, d = 0;
        for (; d < 256; ++d) {
          int h = hist[d];
          if (acc + h > k) break;
          acc += h;
        }
        s_prefix = prefix | ((unsigned)d << shift);
        s_k      = k - acc;
      }
      __syncthreads();
      prefix = s_prefix;
      k      = s_k;
      __syncthreads();
    }
    result[sel] = prefix;
  }
  if (tid == 0) {
    med[b] = 0.5f * (__uint_as_float(result[0]) + __uint_as_float(result[1]));
  }
}

// ---------------------------------------------------------------------------
// Kernel 2: node projection + volt head + graph mean + sec MLP head.
// Tiny (18 MFLOP). One block per graph, thread n = node n.
// ---------------------------------------------------------------------------
__global__ void __launch_bounds__(256)
node_sec_kernel(const float* __restrict__ x,
                const float* __restrict__ nW,  const float* __restrict__ nb,
                const float* __restrict__ vW,  const float* __restrict__ vb,
                const float* __restrict__ sW1, const float* __restrict__ sb1,
                const float* __restrict__ sW2, const float* __restrict__ sb2,
                float* __restrict__ volt_out,  float* __restrict__ sec_out) {
  const int b   = blockIdx.x;
  const int n   = threadIdx.x;          // node index within graph
  const int row = b * NNODES + n;

  __shared__ float s_sum[DNODE];
  __shared__ float s_red[64];
  if (n < DNODE) s_sum[n] = 0.f;
  __syncthreads();

  float xf[9];
#pragma unroll
  for (int f = 0; f < 9; ++f) xf[f] = x[row * 9 + f];

  float volt = 0.f;
  for (int d = 0; d < DNODE; ++d) {
    float h = nb[d];
#pragma unroll
    for (int f = 0; f < 9; ++f) h = fmaf(xf[f], nW[f * DNODE + d], h);
    volt = fmaf(h, vW[d], volt);
    // wave-level reduction, then one LDS atomic per wave per dim
    float hs = h;
#pragma unroll
    for (int off = 16; off > 0; off >>= 1) hs += __shfl_down(hs, off, 32);
    if ((n & 31) == 0) atomicAdd(&s_sum[d], hs);
  }
  volt_out[row] = volt + vb[0];
  __syncthreads();

  // sec head: relu(s_mean @ W1 + b1) @ W2 + b2
  if (n < 64) {
    float t = sb1[n];
    for (int d = 0; d < DNODE; ++d)
      t = fmaf(s_sum[d] * (1.f / NNODES), sW1[d * 64 + n], t);
    s_red[n] = fmaxf(t, 0.f) * sW2[n];
  }
  __syncthreads();
  if (n == 0) {
    float acc = 0.f;
    for (int j = 0; j < 64; ++j) acc += s_red[j];
    sec_out[b] = acc + sb2[0];
  }
}

// ---------------------------------------------------------------------------
// Kernel 3: pair path (the 536MB producer). Per wave: 16 pairs.
//   A (16x4 f32): LN'd features [zn, log(zn+eps), zn^2, 0]
//       lanes 0-15  hold (K=0,K=1) of pair M=lane
//       lanes 16-31 hold (K=2,K=3) of pair M=lane-16      (ISA 32-bit A layout)
//   B (4x16 f32): pair_W column tile, same half-lane K split.
//   All 4 B fragments + biases are loaded NON-divergently and hoisted above
//   the feature math so the 4 v_wmma_f32_16x16x4_f32 issue back-to-back.
//   D layout: VGPR r, lanes 0-15 -> (pair p0+r, chan 16t+lane): 64B-dense
//   rows -> fully coalesced NT stores (output >> 192MB L2, streamed).
// ---------------------------------------------------------------------------
__global__ void __launch_bounds__(256)
pair_kernel(const float* __restrict__ z,   const float* __restrict__ med_ws,
            const float* __restrict__ lng, const float* __restrict__ lnb,
            const float* __restrict__ pW,  const float* __restrict__ pb,
            float* __restrict__ zout) {
  const int tid  = threadIdx.x;
  const int lane = tid & 31;
  const int wave = tid >> 5;
  const int m    = lane & 15;
  const int half = lane >> 4;

  const long long p_block = (long long)blockIdx.x * 128;   // 8 waves * 16 pairs
  const int       b       = (int)(p_block >> 16);          // 65536 pairs/graph
  const long long p0      = p_block + (long long)wave * 16;
  const long long p       = p0 + m;

  // ---- hoisted, non-divergent constant loads (overlap with feature math) ----
  v2f   Bt[4];
  float bias[4];
#pragma unroll
  for (int t = 0; t < 4; ++t) {
    const int   col = t * 16 + m;
    const float w0  = pW[0 * DPAIR + col];
    const float w1  = pW[1 * DPAIR + col];
    const float w2  = pW[2 * DPAIR + col];
    Bt[t].x = half ? w2 : w0;    // lanes 0-15: K=0 ; lanes 16-31: K=2
    Bt[t].y = half ? 0.f : w1;   // lanes 0-15: K=1 ; lanes 16-31: K=3 (pad)
    bias[t] = pb[col];
  }
  const float inv = 1.f / (med_ws[b] + EPS_Z);
  const float g0 = lng[0], g1 = lng[1], g2 = lng[2];
  const float o0 = lnb[0], o1 = lnb[1], o2 = lnb[2];

  // ---- features + LayerNorm(3) for pair p (hi half duplicates lo's pair) ----
  const float zn = z[p] * inv;
  const float f0 = zn;
  const float f1 = logf(zn + EPS_Z);
  const float f2 = zn * zn;
  const float mu = (f0 + f1 + f2) * (1.f / 3.f);
  const float d0 = f0 - mu, d1 = f1 - mu, d2 = f2 - mu;
  const float var = (d0 * d0 + d1 * d1 + d2 * d2) * (1.f / 3.f);
  const float rs  = rsqrtf(var + EPS_LN);
  const float e0  = d0 * rs * g0 + o0;
  const float e1  = d1 * rs * g1 + o1;
  const float e2  = d2 * rs * g2 + o2;

  v2f A;
  A.x = half ? e2 : e0;
  A.y = half ? 0.f : e1;

#pragma unroll
  for (int t = 0; t < 4; ++t) {
    v8f c = {};
    c = __builtin_amdgcn_wmma_f32_16x16x4_f32(
        /*neg_a=*/false, A, /*neg_b=*/false, Bt[t],
        /*c_mod=*/(short)0, c, /*reuse_a=*/false, /*reuse_b=*/false);

    const int col = t * 16 + m;
#pragma unroll
    for (int r = 0; r < 8; ++r) {
      const long long row = p0 + r + 8 * half;          // D: M = r (+8 hi half)
      __builtin_nontemporal_store(c[r] + bias[t], &zout[row * DPAIR + col]);
    }
  }
}

// ---------------------------------------------------------------------------
extern "C" void kernel_launch(void* const* d_in, const int* in_sizes, int n_in,
                              void* d_out, int out_size, void* d_ws, size_t ws_size,
                              hipStream_t stream) {
  const float* x   = (const float*)d_in[0];
  const float* z   = (const float*)d_in[1];
  const float* nW  = (const float*)d_in[2];
  const float* nb  = (const float*)d_in[3];
  const float* lng = (const float*)d_in[4];
  const float* lnb = (const float*)d_in[5];
  const float* pW  = (const float*)d_in[6];
  const float* pb  = (const float*)d_in[7];
  const float* vW  = (const float*)d_in[8];
  const float* vb  = (const float*)d_in[9];
  const float* sW1 = (const float*)d_in[10];
  const float* sb1 = (const float*)d_in[11];
  const float* sW2 = (const float*)d_in[12];
  const float* sb2 = (const float*)d_in[13];

  float* volt_out = (float*)d_out;                         // [B*N]    = 8192
  float* sec_out  = volt_out + BGRAPHS * NNODES;           // [B]      = 32
  float* zpad_out = sec_out + BGRAPHS;                     // [B,N,N,64]

  float* med_ws = (float*)d_ws;                            // 32 floats

  median_kernel<<<BGRAPHS, 256, 0, stream>>>(z, med_ws);

  node_sec_kernel<<<BGRAPHS, 256, 0, stream>>>(
      x, nW, nb, vW, vb, sW1, sb1, sW2, sb2, volt_out, sec_out);

  const int total_pairs = BGRAPHS * NNODES * NNODES;       // 2,097,152
  const int blocks      = total_pairs / 128;               // 16,384
  pair_kernel<<<blocks, 256, 0, stream>>>(z, med_ws, lng, lnb, pW, pb, zpad_out);
}